// GeometricModule_83983790506182
// MI455X (gfx1250) — compile-verified
//
#include <hip/hip_runtime.h>
#include <math.h>

typedef float v2f __attribute__((ext_vector_type(2)));
typedef float v8f __attribute__((ext_vector_type(8)));

constexpr int BB    = 4;
constexpr int NPTS  = 8192;
constexpr int KNN_K = 20;
constexpr int MTOT  = BB * NPTS;        // 32768 points total
constexpr int C1 = 12;                  // 10 features padded to 12 (K%4==0)
constexpr int C2 = 64, C3 = 128, C4 = 256;

// ---- workspace layout (bytes) ----
constexpr size_t OFF_PTS  = 0;                                   // MTOT*3 f32
constexpr size_t OFF_SQ   = OFF_PTS  + (size_t)MTOT * 3 * 4;     // MTOT f32
constexpr size_t OFF_KNN  = OFF_SQ   + (size_t)MTOT * 4;         // MTOT*20 i32
constexpr size_t OFF_COMB = OFF_KNN  + (size_t)MTOT * KNN_K * 4; // MTOT*12 f32
constexpr size_t OFF_H1   = OFF_COMB + (size_t)MTOT * C1 * 4;    // MTOT*64 f32
constexpr size_t OFF_H2   = OFF_H1   + (size_t)MTOT * C2 * 4;    // MTOT*128 f32
constexpr size_t OFF_W1P  = OFF_H2   + (size_t)MTOT * C3 * 4;    // 64*12 f32

// ---------------- kernel 0: mask points, squared norms ----------------
__global__ void mask_sq_kernel(const float* __restrict__ pc,
                               const unsigned char* __restrict__ vis,
                               float* __restrict__ pts, float* __restrict__ sq) {
    int i = blockIdx.x * blockDim.x + threadIdx.x;
    if (i >= MTOT) return;
    float m = vis[i] ? 1.0f : 0.0f;
    float x = pc[3 * i + 0] * m;
    float y = pc[3 * i + 1] * m;
    float z = pc[3 * i + 2] * m;
    pts[3 * i + 0] = x;
    pts[3 * i + 1] = y;
    pts[3 * i + 2] = z;
    sq[i] = x * x + y * y + z * z;
}

// ---------------- kernel 1: KNN, one wave32 per point ----------------
// Successive lexicographic minima of (d, j) where d = |pj|^2 - 2*pi.pj
// (same ordering as full squared distance for a fixed i; stable ties on j,
//  matching jax.lax.top_k). First selection is self (d = -|pi|^2), dropped.
__global__ void knn_kernel(const float* __restrict__ pts,
                           const float* __restrict__ sq,
                           int* __restrict__ knn) {
    const int lane = threadIdx.x & 31;
    const int wave = threadIdx.x >> 5;
    const int gp   = blockIdx.x * 8 + wave;      // global point id
    const int b    = gp / NPTS;
    const float pix = pts[3 * gp + 0];
    const float piy = pts[3 * gp + 1];
    const float piz = pts[3 * gp + 2];
    const float* __restrict__ bp  = pts + (size_t)b * NPTS * 3;
    const float* __restrict__ bsq = sq  + (size_t)b * NPTS;

    float lastd = -INFINITY;
    int   lastj = -1;
    for (int t = 0; t <= KNN_K; ++t) {
        float bd = INFINITY;
        int   bj = NPTS;
        for (int j = lane; j < NPTS; j += 32) {
            float qx = bp[3 * j + 0], qy = bp[3 * j + 1], qz = bp[3 * j + 2];
            float d  = bsq[j] - 2.0f * (pix * qx + piy * qy + piz * qz);
            bool ok     = (d > lastd) || (d == lastd && j > lastj);
            bool better = (d < bd)    || (d == bd    && j < bj);
            if (ok && better) { bd = d; bj = j; }
        }
        // wave32 butterfly lexicographic-min reduction
        for (int off = 16; off > 0; off >>= 1) {
            float od = __shfl_xor(bd, off, 32);
            int   oj = __shfl_xor(bj, off, 32);
            if (od < bd || (od == bd && oj < bj)) { bd = od; bj = oj; }
        }
        lastd = bd; lastj = bj;
        if (t >= 1 && lane == 0) knn[(size_t)gp * KNN_K + (t - 1)] = bj;
    }
}

// ---------------- kernel 2: covariance + analytic 3x3 eigh ----------------
__global__ void geom_kernel(const float* __restrict__ pts,
                            const int* __restrict__ knn,
                            float* __restrict__ comb) {
    int gp = blockIdx.x * blockDim.x + threadIdx.x;
    if (gp >= MTOT) return;
    const int b = gp / NPTS;
    const float px = pts[3 * gp + 0], py = pts[3 * gp + 1], pz = pts[3 * gp + 2];

    float sx = 0, sy = 0, sz = 0;
    float xx = 0, xy = 0, xz = 0, yy = 0, yz = 0, zz = 0;
    for (int k = 0; k < KNN_K; ++k) {
        int j = knn[(size_t)gp * KNN_K + k];
        size_t g = (size_t)(b * NPTS + j) * 3;
        float lx = pts[g + 0] - px;
        float ly = pts[g + 1] - py;
        float lz = pts[g + 2] - pz;
        sx += lx; sy += ly; sz += lz;
        xx += lx * lx; xy += lx * ly; xz += lx * lz;
        yy += ly * ly; yz += ly * lz; zz += lz * lz;
    }
    const float inv_k = 1.0f / (float)KNN_K;
    const float mx = sx * inv_k, my = sy * inv_k, mz = sz * inv_k;
    // cov = sum(l l^T) - K * m m^T
    float a00 = xx - KNN_K * mx * mx;
    float a01 = xy - KNN_K * mx * my;
    float a02 = xz - KNN_K * mx * mz;
    float a11 = yy - KNN_K * my * my;
    float a12 = yz - KNN_K * my * mz;
    float a22 = zz - KNN_K * mz * mz;

    float w0, w1, w2;          // ascending eigenvalues
    float vx, vy, vz;          // eigenvector of w0
    float p1 = a01 * a01 + a02 * a02 + a12 * a12;
    if (p1 < 1e-20f) {
        // already diagonal
        w0 = a00; vx = 1.f; vy = 0.f; vz = 0.f;
        if (a11 < w0) { w0 = a11; vx = 0.f; vy = 1.f; vz = 0.f; }
        if (a22 < w0) { w0 = a22; vx = 0.f; vy = 0.f; vz = 1.f; }
        float hi = fmaxf(a00, fmaxf(a11, a22));
        w2 = hi;
        w1 = a00 + a11 + a22 - w0 - w2;
    } else {
        float q  = (a00 + a11 + a22) * (1.0f / 3.0f);
        float p2 = (a00 - q) * (a00 - q) + (a11 - q) * (a11 - q) +
                   (a22 - q) * (a22 - q) + 2.0f * p1;
        float p   = sqrtf(p2 * (1.0f / 6.0f));
        float ip  = 1.0f / fmaxf(p, 1e-20f);
        float b00 = (a00 - q) * ip, b11 = (a11 - q) * ip, b22 = (a22 - q) * ip;
        float b01 = a01 * ip, b02 = a02 * ip, b12 = a12 * ip;
        float detB = b00 * (b11 * b22 - b12 * b12)
                   - b01 * (b01 * b22 - b12 * b02)
                   + b02 * (b01 * b12 - b11 * b02);
        float r = fminf(1.0f, fmaxf(-1.0f, 0.5f * detB));
        float phi = acosf(r) * (1.0f / 3.0f);
        float e1 = q + 2.0f * p * cosf(phi);                       // largest
        float e3 = q + 2.0f * p * cosf(phi + 2.09439510239319549f); // smallest
        float e2 = 3.0f * q - e1 - e3;
        w0 = e3; w1 = e2; w2 = e1;
        // eigenvector: best cross product of rows of (A - w0 I)
        float m00 = a00 - w0, m11 = a11 - w0, m22 = a22 - w0;
        float c0x = a01 * a12 - a02 * m11, c0y = a02 * a01 - m00 * a12, c0z = m00 * m11 - a01 * a01;
        float c1x = a01 * m22 - a02 * a12, c1y = a02 * a02 - m00 * m22, c1z = m00 * a12 - a01 * a02;
        float c2x = m11 * m22 - a12 * a12, c2y = a12 * a02 - a01 * m22, c2z = a01 * a12 - m11 * a02;
        float n0 = c0x * c0x + c0y * c0y + c0z * c0z;
        float n1 = c1x * c1x + c1y * c1y + c1z * c1z;
        float n2 = c2x * c2x + c2y * c2y + c2z * c2z;
        vx = c0x; vy = c0y; vz = c0z; float nb = n0;
        if (n1 > nb) { vx = c1x; vy = c1y; vz = c1z; nb = n1; }
        if (n2 > nb) { vx = c2x; vy = c2y; vz = c2z; nb = n2; }
    }
    // sign flip: dot(n, -p) < 0 -> negate   (i.e. n.p > 0 -> negate)
    if (vx * px + vy * py + vz * pz > 0.0f) { vx = -vx; vy = -vy; vz = -vz; }
    float nrm = sqrtf(vx * vx + vy * vy + vz * vz);
    float inv = 1.0f / fmaxf(nrm, 1e-12f);
    vx *= inv; vy *= inv; vz *= inv;
    float curv = w0 / (w0 + w1 + w2 + 1e-10f);

    float* o = comb + (size_t)gp * C1;
    o[0] = px;  o[1] = py;  o[2] = pz;
    o[3] = vx;  o[4] = vy;  o[5] = vz;
    o[6] = curv;
    o[7] = mx;  o[8] = my;  o[9] = mz;
    o[10] = 0.0f; o[11] = 0.0f;          // K-padding for WMMA
}

// ---------------- pad W1 (64x10) -> (64x12) ----------------
__global__ void pack_w1_kernel(const float* __restrict__ w1, float* __restrict__ w1p) {
    int i = blockIdx.x * blockDim.x + threadIdx.x;
    if (i >= 64 * C1) return;
    int o = i / C1, c = i % C1;
    w1p[i] = (c < 10) ? w1[o * 10 + c] : 0.0f;
}

// ---------------- MLP layer: fp32 WMMA 16x16x4, one 16x16 tile/wave ----------------
// Y[m][n] = act( sum_k X[m][k] * W[n][k] + bias[n] )
// A 16x4 layout: lane(0..31): m = lane&15, holds k = k0 + 2*(lane>>4) + {0,1}
// B  4x16 layout: lane:        n = lane&15, holds k = k0 + 2*(lane>>4) + {0,1} (B[k][n] = W[n][k])
// C/D: lane col n = lane&15; VGPR r -> row m = 8*(lane>>4) + r
template <int CIN, int COUT, bool RELU, bool TRANS>
__global__ void mlp_wmma_kernel(const float* __restrict__ X,
                                const float* __restrict__ W,
                                const float* __restrict__ bias,
                                float* __restrict__ Y) {
    const int lane   = threadIdx.x & 31;
    const int wave   = threadIdx.x >> 5;
    const int ntiles = COUT / 16;
    const int tile   = blockIdx.x * 8 + wave;
    const int tm     = (tile / ntiles) * 16;
    const int tn     = (tile % ntiles) * 16;
    const int r      = lane & 15;
    const int h      = lane >> 4;

    const float* __restrict__ arow = X + (size_t)(tm + r) * CIN + 2 * h;
    const float* __restrict__ brow = W + (size_t)(tn + r) * CIN + 2 * h;

    v8f c = {};
#pragma unroll
    for (int k0 = 0; k0 < CIN; k0 += 4) {
        v2f a, bb;
        a.x  = arow[k0];  a.y  = arow[k0 + 1];
        bb.x = brow[k0];  bb.y = brow[k0 + 1];
        c = __builtin_amdgcn_wmma_f32_16x16x4_f32(
                /*neg_a=*/false, a, /*neg_b=*/false, bb,
                /*c_mod=*/(short)0, c, /*reuse_a=*/false, /*reuse_b=*/false);
    }

    const float bn = bias[tn + r];
#pragma unroll
    for (int rr = 0; rr < 8; ++rr) {
        float v = c[rr] + bn;
        if (RELU) v = fmaxf(v, 0.0f);
        int m = tm + 8 * h + rr;
        int n = tn + r;
        if (TRANS) {
            int bidx = m / NPTS, pidx = m % NPTS;       // out[b][n][point]
            Y[((size_t)bidx * COUT + n) * NPTS + pidx] = v;
        } else {
            Y[(size_t)m * COUT + n] = v;
        }
    }
}

extern "C" void kernel_launch(void* const* d_in, const int* in_sizes, int n_in,
                              void* d_out, int out_size, void* d_ws, size_t ws_size,
                              hipStream_t stream) {
    const float*         pc  = (const float*)d_in[0];
    const unsigned char* vis = (const unsigned char*)d_in[1];
    const float* W1 = (const float*)d_in[2];
    const float* b1 = (const float*)d_in[3];
    const float* W2 = (const float*)d_in[4];
    const float* b2 = (const float*)d_in[5];
    const float* W3 = (const float*)d_in[6];
    const float* b3 = (const float*)d_in[7];
    float* out = (float*)d_out;

    char* ws = (char*)d_ws;
    float* pts  = (float*)(ws + OFF_PTS);
    float* sq   = (float*)(ws + OFF_SQ);
    int*   knn  = (int*)  (ws + OFF_KNN);
    float* comb = (float*)(ws + OFF_COMB);
    float* h1   = (float*)(ws + OFF_H1);
    float* h2   = (float*)(ws + OFF_H2);
    float* w1p  = (float*)(ws + OFF_W1P);

    mask_sq_kernel<<<MTOT / 256, 256, 0, stream>>>(pc, vis, pts, sq);
    knn_kernel<<<MTOT / 8, 256, 0, stream>>>(pts, sq, knn);
    geom_kernel<<<MTOT / 256, 256, 0, stream>>>(pts, knn, comb);
    pack_w1_kernel<<<3, 256, 0, stream>>>(W1, w1p);

    // (M/16)*(COUT/16) wave-tiles, 8 waves (256 threads) per block
    mlp_wmma_kernel<C1, C2, true,  false>
        <<<(MTOT / 16) * (C2 / 16) / 8, 256, 0, stream>>>(comb, w1p, b1, h1);
    mlp_wmma_kernel<C2, C3, true,  false>
        <<<(MTOT / 16) * (C3 / 16) / 8, 256, 0, stream>>>(h1, W2, b2, h2);
    mlp_wmma_kernel<C3, C4, false, true>
        <<<(MTOT / 16) * (C4 / 16) / 8, 256, 0, stream>>>(h2, W3, b3, out);
}